// Model_16844861735264
// MI455X (gfx1250) — compile-verified
//
#include <hip/hip_runtime.h>

#define N_SRC   100000
#define N_DST   50000
#define N_EDGES 1600000
#define DIM     128
#define NH      4
#define AF      128
#define FEAT    (NH * AF)   // 512
#define NEG_SLOPE 0.2f

#define LDA 132   // padded leading dim for staged A tiles (bank-conflict-free float2 reads)
#define LDB 136   // padded leading dim for staged B tiles (half-wave rows land 16 banks apart)

typedef __attribute__((ext_vector_type(2))) float v2f;
typedef __attribute__((ext_vector_type(8))) float v8f;

__device__ __forceinline__ float lrelu(float x) { return x >= 0.0f ? x : NEG_SLOPE * x; }

__device__ __forceinline__ float wave_sum(float v) {
#pragma unroll
  for (int o = 16; o > 0; o >>= 1) v += __shfl_xor(v, o, 32);
  return v;
}
__device__ __forceinline__ float wave_max(float v) {
#pragma unroll
  for (int o = 16; o > 0; o >>= 1) v = fmaxf(v, __shfl_xor(v, o, 32));
  return v;
}

// float atomic max via signed/unsigned integer atomics (valid for -inf init).
__device__ __forceinline__ void atomic_max_float(float* addr, float val) {
  if (val >= 0.0f) atomicMax((int*)addr, __float_as_int(val));
  else             atomicMin((unsigned int*)addr, __float_as_uint(val));
}

// ---------------------------------------------------------------------------
// CDNA5 async global->LDS copy (16B per lane), tracked on ASYNCcnt.
// Low 32 bits of a generic shared-memory pointer are the LDS byte offset.
// ---------------------------------------------------------------------------
__device__ __forceinline__ void async_g2l_b128(void* lds, const void* gaddr) {
  unsigned l = (unsigned)(uintptr_t)lds;
  unsigned long long g = (unsigned long long)(uintptr_t)gaddr;
  asm volatile("global_load_async_to_lds_b128 %0, %1, off" :: "v"(l), "v"(g) : "memory");
}
__device__ __forceinline__ void wait_async0() {
  asm volatile("s_wait_asynccnt 0x0" ::: "memory");
}

// ---------------------------------------------------------------------------
// Workspace re-init (deterministic across graph replays)
// ---------------------------------------------------------------------------
__global__ void init_ws_kernel(float* __restrict__ e_max,
                               float* __restrict__ denom,
                               float* __restrict__ g_sum) {
  int i = blockIdx.x * blockDim.x + threadIdx.x;
  if (i < N_DST * NH) { e_max[i] = -__builtin_inff(); denom[i] = 0.0f; }
  if (i < N_DST * DIM) g_sum[i] = 0.0f;
}

// ---------------------------------------------------------------------------
// C[M,N] = A[M,128] @ B[128,N] + bias[N]   (fp32 WMMA 16x16x4)
// Block = 256 threads (8 waves) = 1 M-tile x 8 N-tiles.
// A tile (16x128) and B tile (128x128) async-staged into LDS once per block.
// grid.x = (M/16) * (N/128)
// ---------------------------------------------------------------------------
__global__ void __launch_bounds__(256)
gemm_bias_kernel(const float* __restrict__ A,
                 const float* __restrict__ B,
                 const float* __restrict__ bias,
                 float* __restrict__ C,
                 int M, int N) {
  __shared__ float As[16 * LDA];     //  8.25 KB
  __shared__ float Bs[128 * LDB];    // 68 KB
  const int tid  = threadIdx.x;
  const int lane = tid & 31, wave = tid >> 5;
  const int ntg_per_m = N >> 7;                 // N/128 column groups
  const int mt  = blockIdx.x / ntg_per_m;
  const int ntg = blockIdx.x % ntg_per_m;
  const int colbase = ntg << 7;

  // Stage A tile: 2048 floats = 512 x b128, coalesced (32 lanes span one row).
#pragma unroll
  for (int i = 0; i < 2; ++i) {
    int idx = tid + (i << 8);
    int row = idx >> 5, c4 = (idx & 31) << 2;
    async_g2l_b128(&As[row * LDA + c4], A + ((long)(mt * 16 + row) << 7) + c4);
  }
  // Stage B tile: 16384 floats = 4096 x b128.
#pragma unroll
  for (int i = 0; i < 16; ++i) {
    int idx = tid + (i << 8);
    int row = idx >> 5, c4 = (idx & 31) << 2;
    async_g2l_b128(&Bs[row * LDB + c4], B + (long)row * N + colbase + c4);
  }
  wait_async0();
  __syncthreads();

  const int l16 = lane & 15, hf = lane >> 4;
  const int colL = (wave << 4) + l16;
  v8f c = {};
  for (int k0 = 0; k0 < DIM; k0 += 4) {
    const int k = k0 + (hf << 1);     // lanes 0-15: K=k0,k0+1 ; 16-31: k0+2,k0+3
    float2 av = *(const float2*)&As[l16 * LDA + k];
    v2f a; a.x = av.x; a.y = av.y;
    v2f b; b.x = Bs[k * LDB + colL];
           b.y = Bs[(k + 1) * LDB + colL];
    c = __builtin_amdgcn_wmma_f32_16x16x4_f32(false, a, false, b, (short)0, c, false, false);
  }
  const int col = colbase + colL;
  const float bb = bias[col];
  const int mbase = (mt << 4) + (hf << 3);
#pragma unroll
  for (int v = 0; v < 8; ++v)
    C[(long)(mbase + v) * N + col] = c[v] + bb;
}

// ---------------------------------------------------------------------------
// Per-edge attention score: e[e,h] = sum_af lrelu(fs+fd)*attn ; atomic max per (dst,h)
// one wave per edge, float4 per lane covers AF=128. blockDim = 256 (8 waves).
// ---------------------------------------------------------------------------
__global__ void edge_score_kernel(const int* __restrict__ edge_src,
                                  const int* __restrict__ edge_dst,
                                  const float* __restrict__ feat_src,
                                  const float* __restrict__ feat_dst,
                                  const float* __restrict__ attn,
                                  float* __restrict__ e_score,
                                  float* __restrict__ e_max) {
  const int lane = threadIdx.x & 31;
  const int wave = threadIdx.x >> 5;
  const long e = (long)blockIdx.x * 8 + wave;
  const int s = edge_src[e];
  const int d = edge_dst[e];
  const float4* fs = (const float4*)(feat_src + (long)s * FEAT);
  const float4* fd = (const float4*)(feat_dst + (long)d * FEAT);
  const float4* at = (const float4*)attn;
#pragma unroll
  for (int h = 0; h < NH; ++h) {
    float4 a = fs[h * 32 + lane];
    float4 b = fd[h * 32 + lane];
    float4 w = at[h * 32 + lane];
    float p = lrelu(a.x + b.x) * w.x + lrelu(a.y + b.y) * w.y +
              lrelu(a.z + b.z) * w.z + lrelu(a.w + b.w) * w.w;
    float sum = wave_sum(p);
    if (lane == 0) {
      e_score[e * NH + h] = sum;
      atomic_max_float(&e_max[(long)d * NH + h], sum);
    }
  }
}

// ---------------------------------------------------------------------------
// e_exp = exp(e - emax[dst]); denom[dst,h] += e_exp.  Thread per (edge,head).
// ---------------------------------------------------------------------------
__global__ void edge_exp_kernel(const int* __restrict__ edge_dst,
                                float* __restrict__ e_score,
                                const float* __restrict__ e_max,
                                float* __restrict__ denom) {
  const long i = (long)blockIdx.x * blockDim.x + threadIdx.x;  // over E*NH
  const long e = i >> 2;
  const int h = (int)(i & 3);
  const int d = edge_dst[e];
  const float v = __expf(e_score[i] - e_max[(long)d * NH + h]);
  e_score[i] = v;
  atomicAdd(&denom[(long)d * NH + h], v);
}

// ---------------------------------------------------------------------------
// Per-edge message: a = e_exp/denom; msg = softmax_D(h_src*rr*a) per head;
// scatter-add head-summed msg into g_sum[dst, :].  One wave per edge.
// ---------------------------------------------------------------------------
__global__ void message_kernel(const int* __restrict__ edge_src,
                               const int* __restrict__ edge_dst,
                               const int* __restrict__ edge_type,
                               const float* __restrict__ h_src,
                               const float* __restrict__ rel_table,
                               const float* __restrict__ e_score,
                               const float* __restrict__ denom,
                               float* __restrict__ g_sum) {
  const int lane = threadIdx.x & 31;
  const int wave = threadIdx.x >> 5;
  const long e = (long)blockIdx.x * 8 + wave;
  const int s = edge_src[e], d = edge_dst[e], r = edge_type[e];
  const float4 hs = ((const float4*)(h_src + (long)s * DIM))[lane];
  const float4 rr = ((const float4*)(rel_table + (long)r * DIM))[lane];
  const float px = hs.x * rr.x, py = hs.y * rr.y;
  const float pz = hs.z * rr.z, pw = hs.w * rr.w;
  float acc0 = 0.f, acc1 = 0.f, acc2 = 0.f, acc3 = 0.f;
#pragma unroll
  for (int h = 0; h < NH; ++h) {
    const float a = e_score[e * NH + h] / denom[(long)d * NH + h];
    const float x0 = px * a, x1 = py * a, x2 = pz * a, x3 = pw * a;
    const float m = wave_max(fmaxf(fmaxf(x0, x1), fmaxf(x2, x3)));
    const float e0 = __expf(x0 - m), e1 = __expf(x1 - m);
    const float e2 = __expf(x2 - m), e3 = __expf(x3 - m);
    const float inv = 1.0f / wave_sum(e0 + e1 + e2 + e3);
    acc0 += e0 * inv; acc1 += e1 * inv; acc2 += e2 * inv; acc3 += e3 * inv;
  }
  float* g = g_sum + (long)d * DIM + lane * 4;
  atomicAdd(g + 0, acc0);
  atomicAdd(g + 1, acc1);
  atomicAdd(g + 2, acc2);
  atomicAdd(g + 3, acc3);
}

// ---------------------------------------------------------------------------
// out = lrelu((h+g)@W1 + b1) + lrelu((h*g)@W4 + b4)
// Block = 256 threads (8 waves) = 1 M-tile x all 8 N-tiles; both weight
// matrices (64 KB each) async-staged into LDS; dual WMMA per K-step.
// grid.x = N_DST/16 = 3125
// ---------------------------------------------------------------------------
__global__ void __launch_bounds__(256)
final_kernel(const float* __restrict__ h_dst,
             const float* __restrict__ g_sum,
             const float* __restrict__ w1, const float* __restrict__ b1,
             const float* __restrict__ w4, const float* __restrict__ b4,
             float* __restrict__ out) {
  __shared__ float Ws1[128 * LDB];   // 68 KB
  __shared__ float Ws4[128 * LDB];   // 68 KB
  const int tid  = threadIdx.x;
  const int lane = tid & 31, wave = tid >> 5;
  const int mt = blockIdx.x;

#pragma unroll
  for (int i = 0; i < 16; ++i) {
    int idx = tid + (i << 8);
    int row = idx >> 5, c4 = (idx & 31) << 2;
    async_g2l_b128(&Ws1[row * LDB + c4], w1 + ((long)row << 7) + c4);
    async_g2l_b128(&Ws4[row * LDB + c4], w4 + ((long)row << 7) + c4);
  }
  wait_async0();
  __syncthreads();

  const int l16 = lane & 15, hf = lane >> 4;
  const int row  = (mt << 4) + l16;
  const int colL = (wave << 4) + l16;
  const float* hr = h_dst + ((long)row << 7);
  const float* gr = g_sum + ((long)row << 7);

  v8f c1 = {};
  v8f c2 = {};
  for (int k0 = 0; k0 < DIM; k0 += 4) {
    const int k = k0 + (hf << 1);
    float2 hd = *(const float2*)(hr + k);
    float2 gs = *(const float2*)(gr + k);
    v2f a1; a1.x = hd.x + gs.x; a1.y = hd.y + gs.y;   // A for W1 path
    v2f a2; a2.x = hd.x * gs.x; a2.y = hd.y * gs.y;   // A for W4 path
    v2f bw1; bw1.x = Ws1[k * LDB + colL]; bw1.y = Ws1[(k + 1) * LDB + colL];
    v2f bw4; bw4.x = Ws4[k * LDB + colL]; bw4.y = Ws4[(k + 1) * LDB + colL];
    c1 = __builtin_amdgcn_wmma_f32_16x16x4_f32(false, a1, false, bw1, (short)0, c1, false, false);
    c2 = __builtin_amdgcn_wmma_f32_16x16x4_f32(false, a2, false, bw4, (short)0, c2, false, false);
  }
  const float bb1 = b1[colL], bb4 = b4[colL];
  const int mbase = (mt << 4) + (hf << 3);
#pragma unroll
  for (int v = 0; v < 8; ++v) {
    out[(long)(mbase + v) * DIM + colL] = lrelu(c1[v] + bb1) + lrelu(c2[v] + bb4);
  }
}

// ---------------------------------------------------------------------------
extern "C" void kernel_launch(void* const* d_in, const int* in_sizes, int n_in,
                              void* d_out, int out_size, void* d_ws, size_t ws_size,
                              hipStream_t stream) {
  (void)in_sizes; (void)n_in; (void)out_size; (void)ws_size;
  const float* h_src    = (const float*)d_in[0];
  const float* h_dst    = (const float*)d_in[1];
  const int*   edge_src = (const int*)d_in[2];
  const int*   edge_dst = (const int*)d_in[3];
  const int*   edge_typ = (const int*)d_in[4];
  const float* rel_tab  = (const float*)d_in[5];
  const float* w1_w     = (const float*)d_in[6];
  const float* w1_b     = (const float*)d_in[7];
  const float* w2s_w    = (const float*)d_in[8];
  const float* w2s_b    = (const float*)d_in[9];
  const float* w2d_w    = (const float*)d_in[10];
  const float* w2d_b    = (const float*)d_in[11];
  const float* w4_w     = (const float*)d_in[12];
  const float* w4_b     = (const float*)d_in[13];
  const float* attn     = (const float*)d_in[14];
  float* out = (float*)d_out;

  // Workspace layout (floats)
  float* ws       = (float*)d_ws;
  float* feat_src = ws;                                        // N_SRC*512
  float* feat_dst = feat_src + (long)N_SRC * FEAT;             // N_DST*512
  float* e_score  = feat_dst + (long)N_DST * FEAT;             // E*4
  float* e_max    = e_score + (long)N_EDGES * NH;              // N_DST*4
  float* denom    = e_max + (long)N_DST * NH;                  // N_DST*4
  float* g_sum    = denom + (long)N_DST * NH;                  // N_DST*128

  // 1) zero / -inf the reduction buffers
  init_ws_kernel<<<(N_DST * DIM + 255) / 256, 256, 0, stream>>>(e_max, denom, g_sum);

  // 2) projection GEMMs (fp32 WMMA, async-LDS staged tiles)
  // feat_src: 100000x512 -> (100000/16) * (512/128) = 25000 blocks
  gemm_bias_kernel<<<25000, 256, 0, stream>>>(h_src, w2s_w, w2s_b, feat_src, N_SRC, FEAT);
  // feat_dst: 50000x512 -> 12500 blocks
  gemm_bias_kernel<<<12500, 256, 0, stream>>>(h_dst, w2d_w, w2d_b, feat_dst, N_DST, FEAT);

  // 3) edge scores + per-(dst,head) running max
  edge_score_kernel<<<N_EDGES / 8, 256, 0, stream>>>(edge_src, edge_dst, feat_src,
                                                     feat_dst, attn, e_score, e_max);

  // 4) exp + denom
  edge_exp_kernel<<<(N_EDGES * NH) / 256, 256, 0, stream>>>(edge_dst, e_score, e_max, denom);

  // 5) normalized message softmax + scatter-add into g_sum
  message_kernel<<<N_EDGES / 8, 256, 0, stream>>>(edge_src, edge_dst, edge_typ, h_src,
                                                  rel_tab, e_score, denom, g_sum);

  // 6) fused dual GEMM + leaky-relu epilogue -> out [50000 x 128]
  final_kernel<<<3125, 256, 0, stream>>>(h_dst, g_sum, w1_w, w1_b, w4_w, w4_b, out);
}